// PopulationGCN_14980845928730
// MI455X (gfx1250) — compile-verified
//
#include <hip/hip_runtime.h>
#include <hip/hip_bf16.h>

// ---------------------------------------------------------------------------
// ChebConv GCN (K=3) for MI455X / gfx1250.
//   stage 1: degree + symmetric-normalized scaled-Laplacian edge weights
//   stage 2: ChebConv1 (96->64): Tx1 = prop(x); Tx2 = 2*prop(Tx1) - x;
//            h1 = relu(sum_k Txk @ W1[k] + b1)   [fp32 WMMA 16x16x4]
//   stage 3: ChebConv2 (64->64) same recursion on h1 -> h2
//   stage 4: out = h2 @ Wfc + bfc  (N x 2)
// Dense GEMMs use V_WMMA_F32_16X16X4_F32 (fp32 matrix pipe, full precision).
// SpMM is wave-per-edge with coalesced gathers + native f32 global atomics
// (unsafeAtomicAdd -> global_atomic_add_f32, never a CAS loop); the feature
// tables (19MB / 12.8MB) are resident in the 192MB L2, so the whole SpMM is
// L2-latency/atomic-throughput bound, not HBM bound.
// ---------------------------------------------------------------------------

typedef __attribute__((ext_vector_type(2))) float v2f;
typedef __attribute__((ext_vector_type(8))) float v8f;

#define IN_DIM  96
#define HID_DIM 64
#define OUT_DIM 2
#define CHEB_K  3

// Native FP32 global atomic add (global_atomic_add_f32), bypassing HIP's
// safe-atomics CAS fallback.
__device__ __forceinline__ void atomic_add_f32(float* p, float v) {
    unsafeAtomicAdd(p, v);
}

// ---------------------------------------------------------------------------
// Edge-weight normalization
// ---------------------------------------------------------------------------
__global__ void deg_kernel(const int* __restrict__ src, const int* __restrict__ dst,
                           const float* __restrict__ ew, float* __restrict__ deg, int E) {
    int e = blockIdx.x * blockDim.x + threadIdx.x;
    if (e >= E) return;
    int s = src[e];
    float v = (s == dst[e]) ? 0.0f : ew[e];
    atomic_add_f32(&deg[s], v);
}

__global__ void dinv_kernel(float* __restrict__ deg, int N) {
    int i = blockIdx.x * blockDim.x + threadIdx.x;
    if (i >= N) return;
    float d = deg[i];
    deg[i] = (d > 0.0f) ? rsqrtf(d) : 0.0f;   // deg buffer becomes dinv in place
}

__global__ void norm_kernel(const int* __restrict__ src, const int* __restrict__ dst,
                            const float* __restrict__ ew, const float* __restrict__ dinv,
                            float* __restrict__ w, int E) {
    int e = blockIdx.x * blockDim.x + threadIdx.x;
    if (e >= E) return;
    int s = src[e], d = dst[e];
    float v = (s == d) ? 0.0f : ew[e];
    w[e] = -dinv[s] * v * dinv[d];
}

// ---------------------------------------------------------------------------
// SpMM: out[dst[e]] += scale * w[e] * x[src[e]]   (wave-per-edge, grid-stride)
// 32 lanes stripe the DIM-float feature row -> coalesced 128B gathers and
// clustered global_atomic_add_f32 scatters.
// ---------------------------------------------------------------------------
template <int DIM>
__global__ __launch_bounds__(256) void prop_kernel(
    const float* __restrict__ x, const int* __restrict__ src,
    const int* __restrict__ dst, const float* __restrict__ w,
    float scale, float* __restrict__ out, int E) {
    const int lane   = threadIdx.x & 31;
    const int waveId = blockIdx.x * (blockDim.x >> 5) + (threadIdx.x >> 5);
    const int nWaves = gridDim.x * (blockDim.x >> 5);
    for (int e = waveId; e < E; e += nWaves) {
        const float wv = w[e] * scale;
        const int s = src[e], d = dst[e];
        const float* __restrict__ xs = x + (long)s * DIM;
        float* __restrict__ od       = out + (long)d * DIM;
#pragma unroll
        for (int i = 0; i < DIM / 32; ++i) {
            const int c = lane + i * 32;
            atomic_add_f32(&od[c], wv * xs[c]);
        }
    }
}

// Tx2 accumulator seed: y = -x  (then prop with scale=2 yields 2*prop - Tx0)
__global__ void negate_kernel(const float* __restrict__ x, float* __restrict__ y, long n) {
    long i = (long)blockIdx.x * blockDim.x + threadIdx.x;
    if (i < n) y[i] = -x[i];
}

// ---------------------------------------------------------------------------
// Dense Chebyshev GEMM:  out = relu(A0@W[0] + A1@W[1] + A2@W[2] + bias)
// M x KDIM  @  KDIM x 64, fp32, via V_WMMA_F32_16X16X4_F32.
// Block = 128 threads = 4 waves; wave w owns N-tile [16w,16w+16).
// Each wave computes MT=4 M-tiles so the B fragment (weight slice) is reused
// across 4 independent WMMA chains (hides accumulate latency).
//
// VGPR layouts (ISA 7.12.2, wave32):
//   A 16x4 f32: lanes 0-15 M=0..15, v[0]=K+0, v[1]=K+1; lanes 16-31 K+2/K+3.
//   B  4x16 f32: mirrored (rows striped across lanes within a VGPR).
//   C/D 16x16:  VGPR r, lanes 0-15 -> (M=r, N=lane); lanes 16-31 -> (M=r+8).
// ---------------------------------------------------------------------------
template <int KDIM, int MT>
__global__ __launch_bounds__(128) void cheb_gemm_kernel(
    const float* __restrict__ A0, const float* __restrict__ A1,
    const float* __restrict__ A2, const float* __restrict__ W,
    const float* __restrict__ bias, float* __restrict__ out, int M) {
    const int lane   = threadIdx.x & 31;
    const int l16    = lane & 15;
    const int half   = lane >> 4;              // 0: K+0/K+1,M+0..7 | 1: K+2/K+3,M+8..15
    const int nBase  = (threadIdx.x >> 5) * 16;
    const int col    = nBase + l16;            // B column / C column for this lane
    const int mBlock = blockIdx.x * (16 * MT);

    v8f acc[MT];
#pragma unroll
    for (int mt = 0; mt < MT; ++mt) acc[mt] = (v8f){};

#pragma unroll
    for (int t = 0; t < CHEB_K; ++t) {
        const float* __restrict__ A  = (t == 0) ? A0 : (t == 1) ? A1 : A2;
        const float* __restrict__ Wt = W + (long)t * KDIM * HID_DIM;
        for (int k = 0; k < KDIM; k += 4) {
            const int ka = k + half * 2;
            v2f b;
            b.x = Wt[(ka + 0) * HID_DIM + col];
            b.y = Wt[(ka + 1) * HID_DIM + col];
#pragma unroll
            for (int mt = 0; mt < MT; ++mt) {
                int row = mBlock + mt * 16 + l16;
                row = row < M ? row : M - 1;                    // tail clamp (no branch)
                const float* __restrict__ ar = A + (long)row * KDIM + ka;
                v2f a; a.x = ar[0]; a.y = ar[1];                // single b64 load
                acc[mt] = __builtin_amdgcn_wmma_f32_16x16x4_f32(
                    false, a, false, b, (short)0, acc[mt], false, false);
            }
        }
    }

    const float bv = bias[col];
#pragma unroll
    for (int mt = 0; mt < MT; ++mt) {
        const int mOff = mBlock + mt * 16 + half * 8;
#pragma unroll
        for (int r = 0; r < 8; ++r) {
            const int m = mOff + r;
            if (m < M) {
                float v = acc[mt][r] + bv;
                out[(long)m * HID_DIM + col] = v > 0.0f ? v : 0.0f;   // relu
            }
        }
    }
}

// ---------------------------------------------------------------------------
// Final FC: out[n,:] = h[n,:] @ Wfc + bfc  (64 -> 2), one node per thread.
// ---------------------------------------------------------------------------
__global__ void fc_kernel(const float* __restrict__ h, const float* __restrict__ Wfc,
                          const float* __restrict__ bfc, float* __restrict__ out, int N) {
    int n = blockIdx.x * blockDim.x + threadIdx.x;
    if (n >= N) return;
    const float4* __restrict__ hp = (const float4*)(h + (long)n * HID_DIM);
    float o0 = bfc[0], o1 = bfc[1];
#pragma unroll
    for (int i = 0; i < HID_DIM / 4; ++i) {
        float4 v = hp[i];
        int k = i * 4;
        o0 += v.x * Wfc[(k + 0) * 2 + 0] + v.y * Wfc[(k + 1) * 2 + 0] +
              v.z * Wfc[(k + 2) * 2 + 0] + v.w * Wfc[(k + 3) * 2 + 0];
        o1 += v.x * Wfc[(k + 0) * 2 + 1] + v.y * Wfc[(k + 1) * 2 + 1] +
              v.z * Wfc[(k + 2) * 2 + 1] + v.w * Wfc[(k + 3) * 2 + 1];
    }
    out[(long)n * 2 + 0] = o0;
    out[(long)n * 2 + 1] = o1;
}

// ---------------------------------------------------------------------------
extern "C" void kernel_launch(void* const* d_in, const int* in_sizes, int n_in,
                              void* d_out, int out_size, void* d_ws, size_t ws_size,
                              hipStream_t stream) {
    const float* x   = (const float*)d_in[0];   // [N, 96]
    const int*   ei  = (const int*)  d_in[1];   // [2, E]
    const float* ew  = (const float*)d_in[2];   // [E]
    const float* W1  = (const float*)d_in[3];   // [3, 96, 64]
    const float* b1  = (const float*)d_in[4];   // [64]
    const float* W2  = (const float*)d_in[5];   // [3, 64, 64]
    const float* b2  = (const float*)d_in[6];   // [64]
    const float* Wfc = (const float*)d_in[7];   // [64, 2]
    const float* bfc = (const float*)d_in[8];   // [2]
    float* out = (float*)d_out;

    const int N = in_sizes[0] / IN_DIM;         // 50000
    const int E = in_sizes[2];                  // 800000
    const int* src = ei;
    const int* dst = ei + E;

    // ---- workspace layout (floats), with reuse after layer-1 GEMM ----
    float* ws = (float*)d_ws;
    const long nAlign = ((long)N + 63) & ~63L;
    float* deg  = ws;                               // [N]      (becomes dinv)
    float* w    = deg + nAlign;                     // [E]
    float* TxA1 = w + (((long)E + 63) & ~63L);      // [N*96]
    float* TxA2 = TxA1 + (long)N * IN_DIM;          // [N*96]
    float* h1   = TxA2 + (long)N * IN_DIM;          // [N*64]
    float* TxB1 = TxA1;                             // [N*64]  reuse (TxA1 dead)
    float* TxB2 = TxA1 + (long)N * HID_DIM;         // [N*64]  reuse (inside TxA1/2)
    float* h2   = TxB2 + (long)N * HID_DIM;         // [N*64]  reuse (inside TxA2)

    const int T256 = 256;
    const int propBlocks = 2048;                    // grid-stride, 16K waves

    // ---- stage 1: normalized edge weights ----
    hipMemsetAsync(deg, 0, (size_t)N * sizeof(float), stream);
    deg_kernel <<<(E + T256 - 1) / T256, T256, 0, stream>>>(src, dst, ew, deg, E);
    dinv_kernel<<<(N + T256 - 1) / T256, T256, 0, stream>>>(deg, N);
    norm_kernel<<<(E + T256 - 1) / T256, T256, 0, stream>>>(src, dst, ew, deg, w, E);

    // ---- stage 2: ChebConv1 (96 -> 64) ----
    hipMemsetAsync(TxA1, 0, (size_t)N * IN_DIM * sizeof(float), stream);
    prop_kernel<IN_DIM><<<propBlocks, T256, 0, stream>>>(x, src, dst, w, 1.0f, TxA1, E);
    negate_kernel<<<(int)(((long)N * IN_DIM + T256 - 1) / T256), T256, 0, stream>>>(
        x, TxA2, (long)N * IN_DIM);
    prop_kernel<IN_DIM><<<propBlocks, T256, 0, stream>>>(TxA1, src, dst, w, 2.0f, TxA2, E);
    cheb_gemm_kernel<IN_DIM, 4><<<(N + 63) / 64, 128, 0, stream>>>(
        x, TxA1, TxA2, W1, b1, h1, N);

    // ---- stage 3: ChebConv2 (64 -> 64) ----
    hipMemsetAsync(TxB1, 0, (size_t)N * HID_DIM * sizeof(float), stream);
    prop_kernel<HID_DIM><<<propBlocks, T256, 0, stream>>>(h1, src, dst, w, 1.0f, TxB1, E);
    negate_kernel<<<(int)(((long)N * HID_DIM + T256 - 1) / T256), T256, 0, stream>>>(
        h1, TxB2, (long)N * HID_DIM);
    prop_kernel<HID_DIM><<<propBlocks, T256, 0, stream>>>(TxB1, src, dst, w, 2.0f, TxB2, E);
    cheb_gemm_kernel<HID_DIM, 4><<<(N + 63) / 64, 128, 0, stream>>>(
        h1, TxB1, TxB2, W2, b2, h2, N);

    // ---- stage 4: FC head ----
    fc_kernel<<<(N + T256 - 1) / T256, T256, 0, stream>>>(h2, Wfc, bfc, out, N);
}